// ROIHead_TSSEMamba_51737176048191
// MI455X (gfx1250) — compile-verified
//
#include <hip/hip_runtime.h>
#include <math.h>

#define NROI   2048
#define DMODEL 512
#define DINNER 1024
#define DSTATE 16
#define DTRANK 32
#define LSPP   7
#define TSEQ   4
#define HIDDEN 4096
#define ROWS   (NROI*TSEQ)   // 8192

typedef __attribute__((ext_vector_type(2))) float v2f;
typedef __attribute__((ext_vector_type(8))) float v8f;

__device__ __forceinline__ float siluf(float x) { return x / (1.0f + __expf(-x)); }

// ---------------------------------------------------------------------------
// fp32 WMMA GEMM:  C[M,N] = act( A[M,K] * W[N,K]^T + bias ) (+= C)
// Register-blocked: each wave computes a 32x32 tile (2 A frags x 2 B frags ->
// 4 V_WMMA_F32_16X16X4_F32 per k-step against 4 b64 loads). 4 waves / block
// arranged 2x2 -> 64x64 block tile. All M, N in this net divide 64.
// Fragment layouts per CDNA5 ISA:
//   A 16x4 : lanes 0-15 M=lane  {v0:K0, v1:K1}; lanes 16-31 {v0:K2, v1:K3}
//   B 4x16 : lanes 0-15 N=lane  {v0:K0, v1:K1}; lanes 16-31 {v0:K2, v1:K3}
//   C 16x16: v[r] = row (tileM + r + (lane<16?0:8)), col = tileN + (lane&15)
// ---------------------------------------------------------------------------
template<int ACT, bool ACCUM>   // ACT: 0 none, 1 relu, 2 softplus
__global__ __launch_bounds__(128)
void gemm_wmma_f32(const float* __restrict__ A, int lda,
                   const float* __restrict__ W, int ldw,
                   float* __restrict__ C, int ldc,
                   const float* __restrict__ bias, int K)
{
    const int lane  = threadIdx.x & 31;
    const int wave  = threadIdx.x >> 5;
    const int wm    = wave >> 1;                    // 0..1
    const int wn    = wave & 1;                     // 0..1
    const int tileM = blockIdx.y * 64 + wm * 32;
    const int tileN = blockIdx.x * 64 + wn * 32;
    const int half  = lane >> 4;                    // which K pair
    const int l16   = lane & 15;

    const float* Arow0 = A + (size_t)(tileM + l16)      * lda + half * 2;
    const float* Arow1 = A + (size_t)(tileM + 16 + l16) * lda + half * 2;
    const float* Wrow0 = W + (size_t)(tileN + l16)      * ldw + half * 2;
    const float* Wrow1 = W + (size_t)(tileN + 16 + l16) * ldw + half * 2;

    v8f c00 = {}, c01 = {}, c10 = {}, c11 = {};
    for (int k0 = 0; k0 < K; k0 += 4) {
        float2 a0v = *reinterpret_cast<const float2*>(Arow0 + k0);
        float2 a1v = *reinterpret_cast<const float2*>(Arow1 + k0);
        float2 b0v = *reinterpret_cast<const float2*>(Wrow0 + k0);
        float2 b1v = *reinterpret_cast<const float2*>(Wrow1 + k0);
        v2f a0; a0[0] = a0v.x; a0[1] = a0v.y;
        v2f a1; a1[0] = a1v.x; a1[1] = a1v.y;
        v2f b0; b0[0] = b0v.x; b0[1] = b0v.y;
        v2f b1; b1[0] = b1v.x; b1[1] = b1v.y;
        c00 = __builtin_amdgcn_wmma_f32_16x16x4_f32(false, a0, false, b0, (short)0, c00, false, false);
        c01 = __builtin_amdgcn_wmma_f32_16x16x4_f32(false, a0, false, b1, (short)0, c01, false, false);
        c10 = __builtin_amdgcn_wmma_f32_16x16x4_f32(false, a1, false, b0, (short)0, c10, false, false);
        c11 = __builtin_amdgcn_wmma_f32_16x16x4_f32(false, a1, false, b1, (short)0, c11, false, false);
    }

    // epilogue: 4 sub-tiles
    #pragma unroll
    for (int i = 0; i < 2; ++i) {
        #pragma unroll
        for (int j = 0; j < 2; ++j) {
            const v8f& c = (i == 0) ? (j == 0 ? c00 : c01) : (j == 0 ? c10 : c11);
            const int n     = tileN + j * 16 + l16;
            const int mbase = tileM + i * 16 + half * 8;
            const float bval = bias ? bias[n] : 0.0f;
            float* Crow = C + (size_t)mbase * ldc + n;
            #pragma unroll
            for (int r = 0; r < 8; ++r) {
                float v = c[r] + bval;
                if (ACT == 1) v = fmaxf(v, 0.0f);
                if (ACT == 2) v = (v > 20.0f) ? v : log1pf(__expf(v));
                if (ACCUM) v += Crow[(size_t)r * ldc];
                Crow[(size_t)r * ldc] = v;
            }
        }
    }
}

// ---------------------------------------------------------------------------
// Front end: conv_a (512->256,k3,s2,p1,relu), maxpool3s2p1 + conv_p(1x1,relu),
// SE (mean -> 32 relu -> 512 sigmoid), scale, write h[n][t][c] (c: a|p concat)
// One block (256 thr) per ROI.
// ---------------------------------------------------------------------------
__global__ __launch_bounds__(256)
void frontend_kernel(const float* __restrict__ x_flat,
                     const float* __restrict__ wa,    // 256x512x3
                     const float* __restrict__ wp,    // 256x512
                     const float* __restrict__ se_w1, const float* __restrict__ se_b1,
                     const float* __restrict__ se_w2, const float* __restrict__ se_b2,
                     float* __restrict__ H)           // [NROI][4][512]
{
    __shared__ float xs[DMODEL * LSPP];   // 3584
    __shared__ float ybuf[DMODEL * 4];    // concat a|p, [ch][t]
    __shared__ float pooled[DMODEL * 4];
    __shared__ float ymean[DMODEL];
    __shared__ float s1[32];

    const int n = blockIdx.x, tid = threadIdx.x;
    const float* xin = x_flat + (size_t)n * (DMODEL * LSPP);
    for (int i = tid; i < DMODEL * LSPP; i += 256) xs[i] = xin[i];
    __syncthreads();

    // max pool (window 3, stride 2, pad 1)
    for (int i = tid; i < DMODEL * 4; i += 256) {
        int ic = i >> 2, ot = i & 3;
        float m = -1e30f;
        #pragma unroll
        for (int j = 0; j < 3; ++j) {
            int ip = 2 * ot - 1 + j;
            if (ip >= 0 && ip < LSPP) m = fmaxf(m, xs[ic * LSPP + ip]);
        }
        pooled[i] = m;
    }

    // conv_a: one output channel per thread
    {
        const int oc = tid;
        float a0 = 0.f, a1 = 0.f, a2 = 0.f, a3 = 0.f;
        const float* wrow = wa + (size_t)oc * DMODEL * 3;
        for (int ic = 0; ic < DMODEL; ++ic) {
            float w0 = wrow[ic * 3 + 0], w1 = wrow[ic * 3 + 1], w2 = wrow[ic * 3 + 2];
            const float* xr = &xs[ic * LSPP];
            a0 += w1 * xr[0] + w2 * xr[1];                  // ot=0: ip=-1,0,1
            a1 += w0 * xr[1] + w1 * xr[2] + w2 * xr[3];     // ot=1: ip=1,2,3
            a2 += w0 * xr[3] + w1 * xr[4] + w2 * xr[5];     // ot=2: ip=3,4,5
            a3 += w0 * xr[5] + w1 * xr[6];                  // ot=3: ip=5,6,7
        }
        ybuf[oc * 4 + 0] = fmaxf(a0, 0.f);
        ybuf[oc * 4 + 1] = fmaxf(a1, 0.f);
        ybuf[oc * 4 + 2] = fmaxf(a2, 0.f);
        ybuf[oc * 4 + 3] = fmaxf(a3, 0.f);
    }
    __syncthreads();

    // conv_p (1x1 over pooled), channels 256..511
    {
        const int oc = tid;
        float p0 = 0.f, p1 = 0.f, p2 = 0.f, p3 = 0.f;
        const float* wrow = wp + (size_t)oc * DMODEL;
        for (int ic = 0; ic < DMODEL; ++ic) {
            float w = wrow[ic];
            const float* pr = &pooled[ic * 4];
            p0 += w * pr[0]; p1 += w * pr[1]; p2 += w * pr[2]; p3 += w * pr[3];
        }
        ybuf[(256 + oc) * 4 + 0] = fmaxf(p0, 0.f);
        ybuf[(256 + oc) * 4 + 1] = fmaxf(p1, 0.f);
        ybuf[(256 + oc) * 4 + 2] = fmaxf(p2, 0.f);
        ybuf[(256 + oc) * 4 + 3] = fmaxf(p3, 0.f);
    }
    __syncthreads();

    for (int ch = tid; ch < DMODEL; ch += 256)
        ymean[ch] = 0.25f * (ybuf[ch*4] + ybuf[ch*4+1] + ybuf[ch*4+2] + ybuf[ch*4+3]);
    __syncthreads();

    if (tid < 32) {
        float acc = se_b1[tid];
        const float* wrow = se_w1 + (size_t)tid * DMODEL;
        for (int ch = 0; ch < DMODEL; ++ch) acc += wrow[ch] * ymean[ch];
        s1[tid] = fmaxf(acc, 0.f);
    }
    __syncthreads();

    for (int ch = tid; ch < DMODEL; ch += 256) {
        float acc = se_b2[ch];
        const float* wrow = se_w2 + (size_t)ch * 32;
        #pragma unroll
        for (int j = 0; j < 32; ++j) acc += wrow[j] * s1[j];
        float sig = 1.0f / (1.0f + __expf(-acc));
        #pragma unroll
        for (int ot = 0; ot < 4; ++ot)
            H[((size_t)n * 4 + ot) * DMODEL + ch] = ybuf[ch * 4 + ot] * sig;
    }
}

// ---------------------------------------------------------------------------
// LayerNorm over last dim (512). One block (256 thr) per row.
// ---------------------------------------------------------------------------
__global__ __launch_bounds__(256)
void layernorm_kernel(const float* __restrict__ X, float* __restrict__ Y,
                      const float* __restrict__ g, const float* __restrict__ b)
{
    __shared__ float red[256];
    const int row = blockIdx.x, tid = threadIdx.x;
    const float* xr = X + (size_t)row * DMODEL;
    float v0 = xr[tid], v1 = xr[tid + 256];

    red[tid] = v0 + v1; __syncthreads();
    for (int s = 128; s > 0; s >>= 1) { if (tid < s) red[tid] += red[tid + s]; __syncthreads(); }
    float mu = red[0] * (1.0f / DMODEL);
    __syncthreads();

    float d0 = v0 - mu, d1 = v1 - mu;
    red[tid] = d0 * d0 + d1 * d1; __syncthreads();
    for (int s = 128; s > 0; s >>= 1) { if (tid < s) red[tid] += red[tid + s]; __syncthreads(); }
    float rs = rsqrtf(red[0] * (1.0f / DMODEL) + 1e-5f);

    float* yr = Y + (size_t)row * DMODEL;
    yr[tid]       = d0 * rs * g[tid]       + b[tid];
    yr[tid + 256] = d1 * rs * g[tid + 256] + b[tid + 256];
}

// ---------------------------------------------------------------------------
// Causal depthwise conv (k=4, left pad 3) + SiLU, with optional time reversal.
// xi = XZ[:, 0:1024]. One thread per (n, c); grid covers 2048*1024.
// ---------------------------------------------------------------------------
__global__ __launch_bounds__(256)
void dwconv_silu_kernel(const float* __restrict__ XZ,  // rows 8192 x 2048
                        const float* __restrict__ cw,  // 1024 x 4
                        const float* __restrict__ cb,  // 1024
                        float* __restrict__ U, int rev)
{
    const int idx = blockIdx.x * 256 + threadIdx.x;
    const int n = idx >> 10, c = idx & 1023;
    const float* w = cw + (size_t)c * 4;
    const float  b = cb[c];

    float xv[TSEQ];
    #pragma unroll
    for (int t = 0; t < TSEQ; ++t) {
        int ts = rev ? (TSEQ - 1 - t) : t;
        xv[t] = XZ[((size_t)n * TSEQ + ts) * (2 * DINNER) + c];
    }
    float* urow = U + ((size_t)n * TSEQ) * DINNER + c;
    #pragma unroll
    for (int t = 0; t < TSEQ; ++t) {
        float acc = b;
        #pragma unroll
        for (int j = 0; j < 4; ++j) {
            int ti = t - 3 + j;
            if (ti >= 0) acc += w[j] * xv[ti];
        }
        urow[(size_t)t * DINNER] = siluf(acc);
    }
}

// ---------------------------------------------------------------------------
// Selective scan: per (n, c) recurrence over T=4 with 16-wide state.
// B = DBL[:,32:48], C = DBL[:,48:64]. Writes (first) / accumulates into Y.
// ---------------------------------------------------------------------------
__global__ __launch_bounds__(256)
void scan_kernel(const float* __restrict__ U,    // 8192 x 1024
                 const float* __restrict__ DT,   // 8192 x 1024
                 const float* __restrict__ DBL,  // 8192 x 64
                 const float* __restrict__ Alog, // 1024 x 16 (this l,d)
                 const float* __restrict__ Dp,   // 1024
                 float* __restrict__ Y, int rev, int first)
{
    const int idx = blockIdx.x * 256 + threadIdx.x;
    const int n = idx >> 10, c = idx & 1023;

    float A[DSTATE];
    const float* al = Alog + (size_t)c * DSTATE;
    #pragma unroll
    for (int s = 0; s < DSTATE; ++s) A[s] = -__expf(al[s]);
    const float D = Dp[c];

    float h[DSTATE];
    #pragma unroll
    for (int s = 0; s < DSTATE; ++s) h[s] = 0.f;

    for (int t = 0; t < TSEQ; ++t) {
        size_t r = (size_t)n * TSEQ + t;
        float dt = DT[r * DINNER + c];
        float u  = U [r * DINNER + c];
        const float* brow = DBL + r * 64 + DTRANK;
        const float* crow = DBL + r * 64 + DTRANK + DSTATE;
        float du = dt * u, acc = 0.f;
        #pragma unroll
        for (int s = 0; s < DSTATE; ++s) {
            h[s] = __expf(dt * A[s]) * h[s] + du * brow[s];
            acc += h[s] * crow[s];
        }
        float y  = acc + u * D;
        int   to = rev ? (TSEQ - 1 - t) : t;
        float* yp = Y + ((size_t)n * TSEQ + to) * DINNER + c;
        if (first) *yp = y; else *yp += y;
    }
}

// y *= silu(z), z = XZ[:, 1024:2048]
__global__ __launch_bounds__(256)
void gate_kernel(float* __restrict__ Y, const float* __restrict__ XZ)
{
    size_t i = (size_t)blockIdx.x * 256 + threadIdx.x;   // 8192*1024
    size_t r = i >> 10; int c = (int)(i & 1023);
    Y[i] *= siluf(XZ[r * (2 * DINNER) + DINNER + c]);
}

// feat[n][c] = mean over t of HN[n][t][c]
__global__ __launch_bounds__(256)
void seqmean_kernel(const float* __restrict__ HN, float* __restrict__ F)
{
    int i = blockIdx.x * 256 + threadIdx.x;              // 2048*512
    int n = i >> 9, c = i & 511;
    const float* h = HN + ((size_t)n * TSEQ) * DMODEL + c;
    F[i] = 0.25f * (h[0] + h[DMODEL] + h[2 * DMODEL] + h[3 * DMODEL]);
}

// ---------------------------------------------------------------------------
extern "C" void kernel_launch(void* const* d_in, const int* in_sizes, int n_in,
                              void* d_out, int out_size, void* d_ws, size_t ws_size,
                              hipStream_t stream)
{
    (void)in_sizes; (void)n_in; (void)out_size; (void)ws_size;
    const float* x_flat    = (const float*)d_in[0];
    const float* conv_a_w  = (const float*)d_in[1];
    const float* conv_p_w  = (const float*)d_in[2];
    const float* se_w1     = (const float*)d_in[3];
    const float* se_b1     = (const float*)d_in[4];
    const float* se_w2     = (const float*)d_in[5];
    const float* se_b2     = (const float*)d_in[6];
    const float* ln_g      = (const float*)d_in[7];
    const float* ln_b      = (const float*)d_in[8];
    const float* in_proj_w = (const float*)d_in[9];
    const float* conv_w    = (const float*)d_in[10];
    const float* conv_b    = (const float*)d_in[11];
    const float* x_proj_w  = (const float*)d_in[12];
    const float* dt_proj_w = (const float*)d_in[13];
    const float* dt_proj_b = (const float*)d_in[14];
    const float* A_log     = (const float*)d_in[15];
    const float* D_param   = (const float*)d_in[16];
    const float* out_proj_w= (const float*)d_in[17];
    const float* out_ln_g  = (const float*)d_in[18];
    const float* out_ln_b  = (const float*)d_in[19];
    const float* mlp_w1    = (const float*)d_in[20];
    const float* mlp_b1    = (const float*)d_in[21];
    const float* mlp_w2    = (const float*)d_in[22];
    const float* mlp_b2    = (const float*)d_in[23];
    float* out = (float*)d_out;

    float* ws  = (float*)d_ws;
    float* H    = ws;                       // 8192*512   = 4,194,304
    float* HN   = ws +  4194304;            // 8192*512
    float* XZ   = ws +  8388608;            // 8192*2048  = 16,777,216
    float* U    = ws + 25165824;            // 8192*1024  = 8,388,608
    float* DBL  = ws + 33554432;            // 8192*64    = 524,288
    float* DT   = ws + 34078720;            // 8192*1024
    float* Y    = ws + 42467328;            // 8192*1024  (end 50,855,936)
    float* FEAT = H;                        // reuse (H dead after final LN)
    float* Z1   = XZ;                       // reuse (XZ dead after layers)

    auto gemm = [&](const float* A, int lda, const float* W, int ldw,
                    float* C, int ldc, const float* bias,
                    int M, int N, int K, int act, bool accum) {
        dim3 g((unsigned)(N / 64), (unsigned)(M / 64)); dim3 blk(128);
        if (act == 0 && !accum)
            gemm_wmma_f32<0,false><<<g, blk, 0, stream>>>(A, lda, W, ldw, C, ldc, bias, K);
        else if (act == 0)
            gemm_wmma_f32<0,true ><<<g, blk, 0, stream>>>(A, lda, W, ldw, C, ldc, bias, K);
        else if (act == 1)
            gemm_wmma_f32<1,false><<<g, blk, 0, stream>>>(A, lda, W, ldw, C, ldc, bias, K);
        else
            gemm_wmma_f32<2,false><<<g, blk, 0, stream>>>(A, lda, W, ldw, C, ldc, bias, K);
    };

    // Front end -> H (2048 x 4 x 512)
    frontend_kernel<<<NROI, 256, 0, stream>>>(x_flat, conv_a_w, conv_p_w,
                                              se_w1, se_b1, se_w2, se_b2, H);

    const int NC = NROI * DINNER / 256;     // 8192 blocks for per-(n,c) kernels
    for (int l = 0; l < 2; ++l) {
        layernorm_kernel<<<ROWS, 256, 0, stream>>>(H, HN, ln_g + l*DMODEL, ln_b + l*DMODEL);
        // xz = hn @ in_proj^T : 8192 x 2048
        gemm(HN, DMODEL, in_proj_w + (size_t)l*2*DINNER*DMODEL, DMODEL,
             XZ, 2*DINNER, nullptr, ROWS, 2*DINNER, DMODEL, 0, false);

        for (int d = 0; d < 2; ++d) {
            int ld = l * 2 + d;
            dwconv_silu_kernel<<<NC, 256, 0, stream>>>(XZ, conv_w + (size_t)ld*DINNER*4,
                                                       conv_b + (size_t)ld*DINNER, U, d);
            // dbl = u @ x_proj^T : 8192 x 64
            gemm(U, DINNER, x_proj_w + (size_t)ld*(DTRANK+2*DSTATE)*DINNER, DINNER,
                 DBL, 64, nullptr, ROWS, 64, DINNER, 0, false);
            // dt = softplus(dtr @ dt_proj^T + b) : 8192 x 1024  (A = DBL cols 0..31)
            gemm(DBL, 64, dt_proj_w + (size_t)ld*DINNER*DTRANK, DTRANK,
                 DT, DINNER, dt_proj_b + (size_t)ld*DINNER, ROWS, DINNER, DTRANK, 2, false);
            scan_kernel<<<NC, 256, 0, stream>>>(U, DT, DBL,
                                                A_log + (size_t)ld*DINNER*DSTATE,
                                                D_param + (size_t)ld*DINNER,
                                                Y, d, d == 0 ? 1 : 0);
        }
        gate_kernel<<<ROWS*DINNER/256, 256, 0, stream>>>(Y, XZ);
        // h += y @ out_proj^T : 8192 x 512
        gemm(Y, DINNER, out_proj_w + (size_t)l*DMODEL*DINNER, DINNER,
             H, DMODEL, nullptr, ROWS, DMODEL, DINNER, 0, true);
    }

    layernorm_kernel<<<ROWS, 256, 0, stream>>>(H, HN, out_ln_g, out_ln_b);
    seqmean_kernel<<<NROI*DMODEL/256, 256, 0, stream>>>(HN, FEAT);

    // MLP: 2048x512 -> 4096 relu -> 4096 relu
    gemm(FEAT, DMODEL, mlp_w1, DMODEL, Z1, HIDDEN, mlp_b1, NROI, HIDDEN, DMODEL, 1, false);
    gemm(Z1, HIDDEN, mlp_w2, HIDDEN, out, HIDDEN, mlp_b2, NROI, HIDDEN, HIDDEN, 1, false);
}